// CriterionCoordinate_23038204576403
// MI455X (gfx1250) — compile-verified
//
#include <hip/hip_runtime.h>
#include <stdint.h>

// Problem constants from the reference (N_POINTS=32768, SEG_LEN=2048, N_SEG=16).
#define SEG    2048
#define ROWS   64               // rows per block-slice
#define BLOCK  256              // 8 wave32 waves
#define SLICES (SEG / ROWS)     // 32
#define CGRPS  (SEG / 4 / 4)    // 128 column groups of 4 per thread
#define CPYSTRIDE (BLOCK * 16)  // 4096 B advanced per copy round (256 lanes x 16B)

// ---------------------------------------------------------------------------
// CDNA5 async global -> LDS copy (GLOBAL_LOAD_ASYNC_TO_LDS_B128).
// VDST VGPR holds the LDS byte offset; VADDR the 64-bit global address.
// INST_OFFSET is added to BOTH the LDS and global addresses (ISA 08 §4.4), so
// a strided 1:1 copy is one base address + N straight-line copies with
// immediate offsets — no per-copy VALU address math.
// Tracked by ASYNCcnt; completed with s_wait_asynccnt.
// ---------------------------------------------------------------------------
__device__ __forceinline__ unsigned lds_off_of(const void* p) {
    // Generic LDS pointers: low 32 bits are the LDS byte offset.
    return (unsigned)(unsigned long long)(uintptr_t)p;
}

template <int OFF>
__device__ __forceinline__ void async_g2l_b128(unsigned lds_byte_off, const void* gaddr) {
    asm volatile("global_load_async_to_lds_b128 %0, %1, off offset:%2"
                 :
                 : "v"(lds_byte_off), "v"((unsigned long long)(uintptr_t)gaddr), "i"(OFF)
                 : "memory");
}

__device__ __forceinline__ void wait_async_zero() {
    asm volatile("s_wait_asynccnt 0" ::: "memory");
}

// One pair (row, col) update: L1 distances, threshold, masked accumulate.
__device__ __forceinline__ void pair_update(float ppx, float ppy, float ppz,
                                            float gpx, float gpy, float gpz, float mr,
                                            float cpx, float cpy, float cpz,
                                            float cgx, float cgy, float cgz, float cm,
                                            float& loss, float& valid) {
    float dp   = fabsf(ppx - cpx) + fabsf(ppy - cpy) + fabsf(ppz - cpz);
    float dg   = fabsf(gpx - cgx) + fabsf(gpy - cgy) + fabsf(gpz - cgz);
    float diff = fabsf(dg - dp);
    float pm   = (diff < 3.0f) ? (mr * cm) : 0.0f;
    loss  = fmaf(pm, diff, loss);
    valid += pm;
}

// ---------------------------------------------------------------------------
// Kernel 1: edge loss partials. grid = 16 * 32 = 512 blocks.
// Block b: segment b>>5, row slice b&31 (64 rows), all 2048 cols.
// Thread t: row = slice*64 + (t & 63), column group = t >> 6 (512 cols each).
// Columns consumed 4 at a time: 4 pts * 12B = 48B (16B aligned) -> column
// reads are 7x ds_load_b128 per 4 columns (conflict-free wave broadcast).
// ---------------------------------------------------------------------------
__global__ __launch_bounds__(BLOCK) void edge_kernel(const float* __restrict__ pred,
                                                     const float* __restrict__ gt,
                                                     const float* __restrict__ mask,
                                                     float* __restrict__ partial) {
    __shared__ float s_pred[SEG * 3];   // 24 KB (raw contiguous, same as global)
    __shared__ float s_gt[SEG * 3];     // 24 KB
    __shared__ float s_mask[SEG];       //  8 KB
    __shared__ float s_red[2 * BLOCK];  //  2 KB

    const int bid   = blockIdx.x;
    const int seg   = bid >> 5;         // / SLICES
    const int slice = bid & (SLICES - 1);
    const int tid   = threadIdx.x;

    const float* pseg = pred + (size_t)seg * SEG * 3;
    const float* gseg = gt   + (size_t)seg * SEG * 3;
    const float* mseg = mask + (size_t)seg * SEG;

    // ---- stage the segment into LDS: straight-line async B128 copies.
    //      Each lane owns one 16B chunk per round; rounds advance by 4096B
    //      via the instruction's immediate offset (applied to LDS + global).
    {
        const unsigned lp = lds_off_of(s_pred) + tid * 16;
        const unsigned lg = lds_off_of(s_gt)   + tid * 16;
        const unsigned lm = lds_off_of(s_mask) + tid * 16;
        const float* gp = pseg + tid * 4;
        const float* gg = gseg + tid * 4;
        const float* gm = mseg + tid * 4;

        async_g2l_b128<0 * CPYSTRIDE>(lp, gp);   // pred: 24KB in 6 rounds
        async_g2l_b128<1 * CPYSTRIDE>(lp, gp);
        async_g2l_b128<2 * CPYSTRIDE>(lp, gp);
        async_g2l_b128<3 * CPYSTRIDE>(lp, gp);
        async_g2l_b128<4 * CPYSTRIDE>(lp, gp);
        async_g2l_b128<5 * CPYSTRIDE>(lp, gp);

        async_g2l_b128<0 * CPYSTRIDE>(lg, gg);   // gt: 24KB in 6 rounds
        async_g2l_b128<1 * CPYSTRIDE>(lg, gg);
        async_g2l_b128<2 * CPYSTRIDE>(lg, gg);
        async_g2l_b128<3 * CPYSTRIDE>(lg, gg);
        async_g2l_b128<4 * CPYSTRIDE>(lg, gg);
        async_g2l_b128<5 * CPYSTRIDE>(lg, gg);

        async_g2l_b128<0 * CPYSTRIDE>(lm, gm);   // mask: 8KB in 2 rounds
        async_g2l_b128<1 * CPYSTRIDE>(lm, gm);
    }

    // ---- pull row data into registers while the async copies are in flight ----
    const int row = slice * ROWS + (tid & (ROWS - 1));
    const float ppx = pseg[row * 3 + 0];
    const float ppy = pseg[row * 3 + 1];
    const float ppz = pseg[row * 3 + 2];
    const float gpx = gseg[row * 3 + 0];
    const float gpy = gseg[row * 3 + 1];
    const float gpz = gseg[row * 3 + 2];
    const float mr  = mseg[row];

    wait_async_zero();
    __syncthreads();

    // ---- inner loop: 128 groups of 4 columns; same group across a wave =>
    //      conflict-free LDS broadcast of 7 x b128 per group ----
    const float4* sp4 = (const float4*)s_pred;
    const float4* sg4 = (const float4*)s_gt;
    const float4* sm4 = (const float4*)s_mask;

    const int g0 = (tid >> 6) * CGRPS;  // column-group start for this thread
    float lossAcc  = 0.0f;
    float validAcc = 0.0f;
#pragma unroll 2
    for (int g = g0; g < g0 + CGRPS; ++g) {
        float4 p0 = sp4[g * 3 + 0];     // pt0.x pt0.y pt0.z pt1.x
        float4 p1 = sp4[g * 3 + 1];     // pt1.y pt1.z pt2.x pt2.y
        float4 p2 = sp4[g * 3 + 2];     // pt2.z pt3.x pt3.y pt3.z
        float4 q0 = sg4[g * 3 + 0];
        float4 q1 = sg4[g * 3 + 1];
        float4 q2 = sg4[g * 3 + 2];
        float4 m4 = sm4[g];

        pair_update(ppx, ppy, ppz, gpx, gpy, gpz, mr,
                    p0.x, p0.y, p0.z, q0.x, q0.y, q0.z, m4.x, lossAcc, validAcc);
        pair_update(ppx, ppy, ppz, gpx, gpy, gpz, mr,
                    p0.w, p1.x, p1.y, q0.w, q1.x, q1.y, m4.y, lossAcc, validAcc);
        pair_update(ppx, ppy, ppz, gpx, gpy, gpz, mr,
                    p1.z, p1.w, p2.x, q1.z, q1.w, q2.x, m4.z, lossAcc, validAcc);
        pair_update(ppx, ppy, ppz, gpx, gpy, gpz, mr,
                    p2.y, p2.z, p2.w, q2.y, q2.z, q2.w, m4.w, lossAcc, validAcc);
    }

    // ---- block reduction (LDS tree; wave32-safe) ----
    s_red[tid]         = lossAcc;
    s_red[BLOCK + tid] = validAcc;
    __syncthreads();
    for (int s = BLOCK / 2; s > 0; s >>= 1) {
        if (tid < s) {
            s_red[tid]         += s_red[tid + s];
            s_red[BLOCK + tid] += s_red[BLOCK + tid + s];
        }
        __syncthreads();
    }
    if (tid == 0) {
        partial[2 * bid + 0] = s_red[0];
        partial[2 * bid + 1] = s_red[BLOCK];
    }
}

// ---------------------------------------------------------------------------
// Kernel 2: reduce partials + node loss + final combine. One block.
// ---------------------------------------------------------------------------
__global__ __launch_bounds__(BLOCK) void final_kernel(const float* __restrict__ pred,
                                                      const float* __restrict__ gt,
                                                      const float* __restrict__ mask,
                                                      const float* __restrict__ partial,
                                                      int nPartial, int nPoints,
                                                      float* __restrict__ out) {
    __shared__ float s_red[4 * BLOCK];
    const int tid = threadIdx.x;

    float e = 0.0f, v = 0.0f;
    for (int i = tid; i < nPartial; i += BLOCK) {
        e += partial[2 * i + 0];
        v += partial[2 * i + 1];
    }

    float nsum = 0.0f, msum = 0.0f;
    for (int i = tid; i < nPoints; i += BLOCK) {
        float m = mask[i];
        float d = fabsf(pred[3 * i + 0] - gt[3 * i + 0]) +
                  fabsf(pred[3 * i + 1] - gt[3 * i + 1]) +
                  fabsf(pred[3 * i + 2] - gt[3 * i + 2]);
        nsum = fmaf(m, d, nsum);
        msum += m;
    }

    s_red[tid]             = e;
    s_red[BLOCK + tid]     = v;
    s_red[2 * BLOCK + tid] = nsum;
    s_red[3 * BLOCK + tid] = msum;
    __syncthreads();
    for (int s = BLOCK / 2; s > 0; s >>= 1) {
        if (tid < s) {
            s_red[tid]             += s_red[tid + s];
            s_red[BLOCK + tid]     += s_red[BLOCK + tid + s];
            s_red[2 * BLOCK + tid] += s_red[2 * BLOCK + tid + s];
            s_red[3 * BLOCK + tid] += s_red[3 * BLOCK + tid + s];
        }
        __syncthreads();
    }

    if (tid == 0) {
        float edgeLoss = s_red[0];
        float valid    = s_red[BLOCK];
        float nodeNum  = s_red[2 * BLOCK];
        float maskSum  = s_red[3 * BLOCK];
        float node     = (nodeNum + 1e-9f) / (maskSum + 1e-9f);
        float edgeTerm = edgeLoss / fmaxf(valid, 1e-9f);
        out[0] = (valid >= 1.0f) ? (node + edgeTerm) : node;
    }
}

// ---------------------------------------------------------------------------
// Launch
// ---------------------------------------------------------------------------
extern "C" void kernel_launch(void* const* d_in, const int* in_sizes, int n_in,
                              void* d_out, int out_size, void* d_ws, size_t ws_size,
                              hipStream_t stream) {
    const float* pred = (const float*)d_in[0];   // [N,3] f32
    const float* gt   = (const float*)d_in[1];   // [N,3] f32
    const float* mask = (const float*)d_in[2];   // [N,1] f32
    // d_in[3] = segment boundary index (int64); uniform segments per reference
    float* out     = (float*)d_out;
    float* partial = (float*)d_ws;               // 2 floats per block

    const int nPoints = in_sizes[0] / 3;         // 32768
    const int nSeg    = nPoints / SEG;           // 16
    const int nBlocks = nSeg * SLICES;           // 512

    edge_kernel<<<nBlocks, BLOCK, 0, stream>>>(pred, gt, mask, partial);
    final_kernel<<<1, BLOCK, 0, stream>>>(pred, gt, mask, partial, nBlocks, nPoints, out);
}